// MergeNN_38903813767173
// MI455X (gfx1250) — compile-verified
//
#include <hip/hip_runtime.h>

// MI455X / gfx1250, wave32. Fused kernel-regression forward.
// Phase A: per 16-query tile, stream all N star/feature rows through LDS,
//   v_wmma_f32_16x16x4_f32 for x.star^T and e^T.feats{1,2}; match tracking,
//   linear head + argmin over L label prototypes.
// Phase B: per 16-query tile, v_wmma for xt.feats^T and e2^T.star_labels.
// Chunk staging uses GLOBAL_LOAD_ASYNC_TO_LDS_B128 (ASYNCcnt) when the
// toolchain exposes the builtin; falls back to float4 staging otherwise.

#define DD 64
#define DYY 32
#define LL 100
#define NW 4

typedef float v2f __attribute__((ext_vector_type(2)));
typedef float v8f __attribute__((ext_vector_type(8)));
typedef int v4i __attribute__((ext_vector_type(4)));

#if defined(__has_builtin)
#if __has_builtin(__builtin_amdgcn_global_load_async_to_lds_b128)
#define HAVE_ASYNC_LDS 1
#endif
#endif

#ifdef HAVE_ASYNC_LDS
// Builtin signature (from clang diagnostic): param0 = v4i in AS(1) (global),
// param1 = v4i in AS(3) (LDS), then imm offset + imm cpol. The same offset is
// added to both the global and the LDS address (ISA 15.18.3).
typedef __attribute__((address_space(1))) v4i ga_v4i;
typedef __attribute__((address_space(3))) v4i ls_v4i;
#define ASYNC_LD_B128(gbase, lbase, off)                                       \
  __builtin_amdgcn_global_load_async_to_lds_b128((ga_v4i*)(gbase),             \
                                                 (ls_v4i*)(lbase), (off), 0)
#define WAIT_ASYNCCNT0() asm volatile("s_wait_asynccnt 0x0" ::: "memory")
#define WAIT_DSCNT0() asm volatile("s_wait_dscnt 0x0" ::: "memory")
#endif

static __device__ inline v8f wmma4(v2f a, v2f b, v8f c) {
  // D = A(16x4, f32) * B(4x16, f32) + C(16x16, f32)
  return __builtin_amdgcn_wmma_f32_16x16x4_f32(false, a, false, b, (short)0, c,
                                               false, false);
}

// A-operand fragment from a row-major [16][stride] tile: lane m = lane&15,
// VGPR v holds K = k0 + (lane>=16 ? 2 : 0) + v.
static __device__ inline v2f frag_mk(const float* T, int stride, int lane, int k0) {
  const int m = lane & 15;
  const int kk = k0 + ((lane & 16) >> 3);
  v2f r = {T[m * stride + kk], T[m * stride + kk + 1]};
  return r;
}

// B-operand fragment (K x N) from a row-major [K][stride] tile: lane n = n0+(lane&15),
// VGPR v holds K-row k0 + (lane>=16 ? 2 : 0) + v.
static __device__ inline v2f frag_kn(const float* T, int stride, int lane, int k0, int n0) {
  const int n = n0 + (lane & 15);
  const int kk = k0 + ((lane & 16) >> 3);
  v2f r = {T[kk * stride + n], T[(kk + 1) * stride + n]};
  return r;
}

static __device__ inline float sq_row64(const float* row) {
  const float4* r = (const float4*)row;
  float s = 0.0f;
#pragma unroll
  for (int i = 0; i < DD / 4; ++i) {
    float4 v = r[i];
    s += v.x * v.x + v.y * v.y + v.z * v.z + v.w * v.w;
  }
  return s;
}

__global__ __launch_bounds__(128) void phaseA_kernel(
    const float* __restrict__ x, const float* __restrict__ star,
    const float* __restrict__ f1, const float* __restrict__ f2,
    const float* __restrict__ W1, const float* __restrict__ b1v,
    const float* __restrict__ W2, const float* __restrict__ b2v,
    const float* __restrict__ u1, const float* __restrict__ u2,
    float* __restrict__ xt1o, float* __restrict__ xt2o,
    int* __restrict__ yi1o, int* __restrict__ yi2o, int N) {
  __shared__ __align__(16) float Xs[16 * DD];            // query tile
  __shared__ __align__(16) float Stg[NW][3][16 * DD];    // per-wave star/f1/f2 chunks
  __shared__ __align__(16) float Els[NW][16 * 16];       // per-wave e-tile (b x n)
  __shared__ float xsq[16];
  __shared__ float esumA[16];
  __shared__ int matchA[16];

  const int tid = threadIdx.x;
  const int lane = tid & 31;
  const int wid = tid >> 5;
  const int b0 = blockIdx.x * 16;
  const int l15 = lane & 15;
  const int mo = (lane & 16) >> 1;  // +8 row offset for hi half-wave (C layout)

  {
    const float4* src = (const float4*)(x + (size_t)b0 * DD);
    float4* dst = (float4*)Xs;
    for (int i = tid; i < 16 * DD / 4; i += 128) dst[i] = src[i];
  }
  if (tid < 16) {
    esumA[tid] = 0.0f;
    matchA[tid] = 0x7fffffff;
  }
  __syncthreads();
  if (tid < 16) xsq[tid] = sq_row64(&Xs[tid * DD]);
  __syncthreads();

  v8f ax1[4], ax2[4];  // xt numerator accumulators: 16b x 64d per branch
  float es[8];
#pragma unroll
  for (int t = 0; t < 4; ++t)
#pragma unroll
    for (int j = 0; j < 8; ++j) {
      ax1[t][j] = 0.0f;
      ax2[t][j] = 0.0f;
    }
#pragma unroll
  for (int j = 0; j < 8; ++j) es[j] = 0.0f;

  float* Ss = Stg[wid][0];
  float* F1s = Stg[wid][1];
  float* F2s = Stg[wid][2];
  float* Es = Els[wid];

  const int nChunks = N / 16;
  for (int c = wid; c < nChunks; c += NW) {
    const int n0 = c * 16;
#ifdef HAVE_ASYNC_LDS
    {
      // WAR guard: previous iteration's ds reads of the staging buffers must
      // retire before the async DMA overwrites them (ASYNC vs DS unordered).
      WAIT_DSCNT0();
      const char* gs = (const char*)(star + (size_t)n0 * DD) + lane * 16;
      const char* ga = (const char*)(f1 + (size_t)n0 * DD) + lane * 16;
      const char* gc = (const char*)(f2 + (size_t)n0 * DD) + lane * 16;
      char* ls = (char*)Ss + lane * 16;
      char* la = (char*)F1s + lane * 16;
      char* lc = (char*)F2s + lane * 16;
#pragma unroll
      for (int k = 0; k < 8; ++k) {  // 8 x 512B per 4KB buffer
        ASYNC_LD_B128(gs + k * 512, ls + k * 512, 0);
        ASYNC_LD_B128(ga + k * 512, la + k * 512, 0);
        ASYNC_LD_B128(gc + k * 512, lc + k * 512, 0);
      }
      WAIT_ASYNCCNT0();
    }
#else
    {
      const float4* s4 = (const float4*)(star + (size_t)n0 * DD);
      const float4* a4 = (const float4*)(f1 + (size_t)n0 * DD);
      const float4* c4 = (const float4*)(f2 + (size_t)n0 * DD);
      float4* sd = (float4*)Ss;
      float4* ad = (float4*)F1s;
      float4* cd4 = (float4*)F2s;
#pragma unroll
      for (int i = lane; i < 16 * DD / 4; i += 32) {
        sd[i] = s4[i];
        ad[i] = a4[i];
        cd4[i] = c4[i];
      }
    }
#endif
    if (c + NW < nChunks) {  // global_prefetch_b8 for the next chunk of this wave
      __builtin_prefetch(star + (size_t)(n0 + NW * 16) * DD, 0, 1);
      __builtin_prefetch(f1 + (size_t)(n0 + NW * 16) * DD, 0, 1);
      __builtin_prefetch(f2 + (size_t)(n0 + NW * 16) * DD, 0, 1);
    }

    const float ssq = sq_row64(&Ss[l15 * DD]);

    // dot[b,n] = sum_d x[b,d]*star[n,d]  (M=b, N=n, K=d=64 -> 16 WMMAs)
    v8f cd;
#pragma unroll
    for (int j = 0; j < 8; ++j) cd[j] = 0.0f;
#pragma unroll
    for (int k0 = 0; k0 < DD; k0 += 4)
      cd = wmma4(frag_mk(Xs, DD, lane, k0), frag_mk(Ss, DD, lane, k0), cd);

    const int ng = n0 + l15;
#pragma unroll
    for (int j = 0; j < 8; ++j) {
      const int m = j + mo;
      float dsq = fmaxf(xsq[m] + ssq - 2.0f * cd[j], 0.0f);
      if (dsq <= 0.0f) atomicMin(&matchA[m], ng);  // first exact match = min n
      const float e = __expf(-dsq);
      es[j] += e;
      Es[m * 16 + l15] = e;  // transpose e-tile through LDS for the A operand
    }

    // acc_xt[b,d] += E(16b x 16n) * F(16n x 64d) : 4 k-steps x 4 d-blocks x 2 branches
#pragma unroll
    for (int t = 0; t < 4; ++t) {
#pragma unroll
      for (int k0 = 0; k0 < 16; k0 += 4) {
        v2f a = frag_mk(Es, 16, lane, k0);
        ax1[t] = wmma4(a, frag_kn(F1s, DD, lane, k0, t * 16), ax1[t]);
        ax2[t] = wmma4(a, frag_kn(F2s, DD, lane, k0, t * 16), ax2[t]);
      }
    }
  }

  __syncthreads();
  // Alias the (now idle) staging area for the cross-wave reduction buffers.
  float* accX1 = &Stg[0][0][0];
  float* accX2 = accX1 + 16 * DD;
  for (int i = tid; i < 2 * 16 * DD; i += 128) accX1[i] = 0.0f;
  __syncthreads();
#pragma unroll
  for (int j = 0; j < 8; ++j) atomicAdd(&esumA[j + mo], es[j]);
#pragma unroll
  for (int t = 0; t < 4; ++t)
#pragma unroll
    for (int j = 0; j < 8; ++j) {
      const int m = j + mo;
      const int d = t * 16 + l15;
      atomicAdd(&accX1[m * DD + d], ax1[t][j]);
      atomicAdd(&accX2[m * DD + d], ax2[t][j]);
    }
  __syncthreads();

  // xt = has_match ? feats[match_idx] : acc/esum ; also spill to workspace
  for (int i = tid; i < 16 * DD; i += 128) {
    const int m = i >> 6, d = i & 63;
    const int mi = matchA[m];
    float v1, v2;
    if (mi != 0x7fffffff) {
      v1 = f1[(size_t)mi * DD + d];
      v2 = f2[(size_t)mi * DD + d];
    } else {
      const float inv = 1.0f / esumA[m];
      v1 = accX1[i] * inv;
      v2 = accX2[i] * inv;
    }
    accX1[i] = v1;
    accX2[i] = v2;
    xt1o[(size_t)(b0 + m) * DD + d] = v1;
    xt2o[(size_t)(b0 + m) * DD + d] = v2;
  }
  __syncthreads();

  // y = xt @ W + b (16x32 per branch), then argmin over L=100 prototypes
  float* yL = &Els[0][0];  // alias: 2 * 16 * 32 = 1024 floats
  for (int i = tid; i < 2 * 16 * DYY; i += 128) {
    const int br = i >= 16 * DYY;
    const int r = i - br * 16 * DYY;
    const int m = r >> 5, dy = r & 31;
    const float* W = br ? W2 : W1;
    const float* bb = br ? b2v : b1v;
    const float* xt = br ? accX2 : accX1;
    float acc = bb[dy];
    for (int d = 0; d < DD; ++d) acc = fmaf(xt[m * DD + d], W[d * DYY + dy], acc);
    yL[i] = acc;
  }
  __syncthreads();
  if (tid < 32) {
    const int br = tid >> 4, m = tid & 15;
    const float* uniq = br ? u2 : u1;
    const float* y = &yL[br * 16 * DYY + m * DYY];
    float best = 3.4e38f;
    int bi = 0;
    for (int l = 0; l < LL; ++l) {
      float s = 0.0f;
      for (int dy = 0; dy < DYY; ++dy) {
        const float df = y[dy] - uniq[l * DYY + dy];
        s = fmaf(df, df, s);
      }
      if (s < best) {
        best = s;
        bi = l;
      }
    }
    (br ? yi2o : yi1o)[b0 + m] = bi;
  }
}

__global__ __launch_bounds__(128) void phaseB_kernel(
    const float* __restrict__ f1, const float* __restrict__ f2,
    const float* __restrict__ slab, const float* __restrict__ ld1,
    const float* __restrict__ ld2, const int* __restrict__ li1,
    const int* __restrict__ li2, const float* __restrict__ xt1,
    const float* __restrict__ xt2, const int* __restrict__ yi1,
    const int* __restrict__ yi2, float* __restrict__ out, int N) {
  __shared__ __align__(16) float XT1[16 * DD];
  __shared__ __align__(16) float XT2[16 * DD];
  __shared__ __align__(16) float Stg[NW][2 * 16 * DD + 16 * DYY];  // f1,f2,star_labels
  __shared__ __align__(16) float Els[NW][16 * 16];
  __shared__ float xtsq1[16], xtsq2[16];
  __shared__ int yiL[32];
  __shared__ float den1A[16], den2A[16];

  const int tid = threadIdx.x;
  const int lane = tid & 31;
  const int wid = tid >> 5;
  const int b0 = blockIdx.x * 16;
  const int l15 = lane & 15;
  const int mo = (lane & 16) >> 1;

  {
    const float4* s1 = (const float4*)(xt1 + (size_t)b0 * DD);
    const float4* s2 = (const float4*)(xt2 + (size_t)b0 * DD);
    float4* d1 = (float4*)XT1;
    float4* d2 = (float4*)XT2;
    for (int i = tid; i < 16 * DD / 4; i += 128) {
      d1[i] = s1[i];
      d2[i] = s2[i];
    }
  }
  if (tid < 32) yiL[tid] = (tid < 16) ? yi1[b0 + tid] : yi2[b0 + (tid - 16)];
  if (tid < 16) {
    den1A[tid] = 0.0f;
    den2A[tid] = 0.0f;
  }
  __syncthreads();
  if (tid < 32) {
    const int br = tid >> 4, m = tid & 15;
    (br ? xtsq2 : xtsq1)[m] = sq_row64(br ? &XT2[m * DD] : &XT1[m * DD]);
  }
  __syncthreads();

  v8f an1[2], an2[2];  // numerator accumulators: 16b x 32dy per branch
  float de1[8], de2[8];
#pragma unroll
  for (int t = 0; t < 2; ++t)
#pragma unroll
    for (int j = 0; j < 8; ++j) {
      an1[t][j] = 0.0f;
      an2[t][j] = 0.0f;
    }
#pragma unroll
  for (int j = 0; j < 8; ++j) {
    de1[j] = 0.0f;
    de2[j] = 0.0f;
  }

  float* F1s = &Stg[wid][0];
  float* F2s = F1s + 16 * DD;
  float* SLs = F2s + 16 * DD;
  float* Es = Els[wid];

  const int nChunks = N / 16;
  for (int c = wid; c < nChunks; c += NW) {
    const int n0 = c * 16;
#ifdef HAVE_ASYNC_LDS
    {
      WAIT_DSCNT0();  // WAR guard vs previous iteration's ds reads
      const char* ga = (const char*)(f1 + (size_t)n0 * DD) + lane * 16;
      const char* gc = (const char*)(f2 + (size_t)n0 * DD) + lane * 16;
      const char* gl = (const char*)(slab + (size_t)n0 * DYY) + lane * 16;
      char* la = (char*)F1s + lane * 16;
      char* lc = (char*)F2s + lane * 16;
      char* ll = (char*)SLs + lane * 16;
#pragma unroll
      for (int k = 0; k < 8; ++k) {
        ASYNC_LD_B128(ga + k * 512, la + k * 512, 0);
        ASYNC_LD_B128(gc + k * 512, lc + k * 512, 0);
      }
#pragma unroll
      for (int k = 0; k < 4; ++k) ASYNC_LD_B128(gl + k * 512, ll + k * 512, 0);
      WAIT_ASYNCCNT0();
    }
#else
    {
      const float4* a4 = (const float4*)(f1 + (size_t)n0 * DD);
      const float4* c4 = (const float4*)(f2 + (size_t)n0 * DD);
      float4* ad = (float4*)F1s;
      float4* cd4 = (float4*)F2s;
#pragma unroll
      for (int i = lane; i < 16 * DD / 4; i += 32) {
        ad[i] = a4[i];
        cd4[i] = c4[i];
      }
      const float4* l4 = (const float4*)(slab + (size_t)n0 * DYY);
      float4* ldst = (float4*)SLs;
#pragma unroll
      for (int i = lane; i < 16 * DYY / 4; i += 32) ldst[i] = l4[i];
    }
#endif
    if (c + NW < nChunks) {
      __builtin_prefetch(f1 + (size_t)(n0 + NW * 16) * DD, 0, 1);
      __builtin_prefetch(f2 + (size_t)(n0 + NW * 16) * DD, 0, 1);
      __builtin_prefetch(slab + (size_t)(n0 + NW * 16) * DYY, 0, 1);
    }
    const int ng = n0 + l15;
    const int r1 = li1[ng], r2 = li2[ng];
    const float* lr1 = ld1 + (size_t)r1 * LL;
    const float* lr2 = ld2 + (size_t)r2 * LL;

    // ---- branch 1 ----
    {
      v8f cd;
#pragma unroll
      for (int j = 0; j < 8; ++j) cd[j] = 0.0f;
#pragma unroll
      for (int k0 = 0; k0 < DD; k0 += 4)
        cd = wmma4(frag_mk(XT1, DD, lane, k0), frag_mk(F1s, DD, lane, k0), cd);
      const float fsq = sq_row64(&F1s[l15 * DD]);
#pragma unroll
      for (int j = 0; j < 8; ++j) {
        const int m = j + mo;
        const float dsq = fmaxf(xtsq1[m] + fsq - 2.0f * cd[j], 0.0f);
        const float e = __expf(-dsq - 0.01f * lr1[yiL[m]]);
        de1[j] += e;
        Es[m * 16 + l15] = e;
      }
#pragma unroll
      for (int t = 0; t < 2; ++t)
#pragma unroll
        for (int k0 = 0; k0 < 16; k0 += 4)
          an1[t] = wmma4(frag_mk(Es, 16, lane, k0),
                         frag_kn(SLs, DYY, lane, k0, t * 16), an1[t]);
    }
    // ---- branch 2 ----
    {
      v8f cd;
#pragma unroll
      for (int j = 0; j < 8; ++j) cd[j] = 0.0f;
#pragma unroll
      for (int k0 = 0; k0 < DD; k0 += 4)
        cd = wmma4(frag_mk(XT2, DD, lane, k0), frag_mk(F2s, DD, lane, k0), cd);
      const float fsq = sq_row64(&F2s[l15 * DD]);
#pragma unroll
      for (int j = 0; j < 8; ++j) {
        const int m = j + mo;
        const float dsq = fmaxf(xtsq2[m] + fsq - 2.0f * cd[j], 0.0f);
        const float e = __expf(-dsq - 0.01f * lr2[yiL[16 + m]]);
        de2[j] += e;
        Es[m * 16 + l15] = e;
      }
#pragma unroll
      for (int t = 0; t < 2; ++t)
#pragma unroll
        for (int k0 = 0; k0 < 16; k0 += 4)
          an2[t] = wmma4(frag_mk(Es, 16, lane, k0),
                         frag_kn(SLs, DYY, lane, k0, t * 16), an2[t]);
    }
  }

  __syncthreads();
  float* num1A = &Stg[0][0];  // alias idle staging for reduction
  float* num2A = num1A + 16 * DYY;
  for (int i = tid; i < 2 * 16 * DYY; i += 128) num1A[i] = 0.0f;
  __syncthreads();
#pragma unroll
  for (int j = 0; j < 8; ++j) {
    atomicAdd(&den1A[j + mo], de1[j]);
    atomicAdd(&den2A[j + mo], de2[j]);
  }
#pragma unroll
  for (int t = 0; t < 2; ++t)
#pragma unroll
    for (int j = 0; j < 8; ++j) {
      const int m = j + mo, dy = t * 16 + l15;
      atomicAdd(&num1A[m * DYY + dy], an1[t][j]);
      atomicAdd(&num2A[m * DYY + dy], an2[t][j]);
    }
  __syncthreads();
  for (int i = tid; i < 16 * DYY; i += 128) {
    const int m = i >> 5;
    out[(size_t)(b0 + m) * DYY + (i & 31)] =
        0.5f * (num1A[i] / den1A[m] + num2A[i] / den2A[m]);
  }
}

extern "C" void kernel_launch(void* const* d_in, const int* in_sizes, int n_in,
                              void* d_out, int out_size, void* d_ws,
                              size_t ws_size, hipStream_t stream) {
  const float* x = (const float*)d_in[0];
  const float* star = (const float*)d_in[1];
  const float* slab = (const float*)d_in[2];
  const float* f1 = (const float*)d_in[3];
  const float* f2 = (const float*)d_in[4];
  const float* u1 = (const float*)d_in[5];
  const float* u2 = (const float*)d_in[6];
  const float* ld1 = (const float*)d_in[7];
  const float* ld2 = (const float*)d_in[8];
  const float* W1 = (const float*)d_in[9];
  const float* b1 = (const float*)d_in[10];
  const float* W2 = (const float*)d_in[11];
  const float* b2 = (const float*)d_in[12];
  const int* li1 = (const int*)d_in[13];
  const int* li2 = (const int*)d_in[14];

  const int B = in_sizes[0] / DD;  // 2048
  const int N = in_sizes[1] / DD;  // 20000

  // Workspace: xt1[B*64], xt2[B*64] f32 + yidx1[B], yidx2[B] i32 (~1.05 MB).
  float* ws = (float*)d_ws;
  float* xt1 = ws;
  float* xt2 = ws + (size_t)B * DD;
  int* yi1 = (int*)(ws + 2 * (size_t)B * DD);
  int* yi2 = yi1 + B;

  dim3 grid(B / 16), block(128);
  phaseA_kernel<<<grid, block, 0, stream>>>(x, star, f1, f2, W1, b1, W2, b2, u1,
                                            u2, xt1, xt2, yi1, yi2, N);
  phaseB_kernel<<<grid, block, 0, stream>>>(f1, f2, slab, ld1, ld2, li1, li2,
                                            xt1, xt2, yi1, yi2, (float*)d_out,
                                            N);
}